// AE_layer_32710470926638
// MI455X (gfx1250) — compile-verified
//
#include <hip/hip_runtime.h>
#include <hip/hip_bf16.h>

#define NN   100000   // nodes
#define FIN  128      // in features
#define HC   256      // H*C
#define CC   64       // out features per head
#define HH   4        // heads
#define EE   800000   // raw edges
#define EF   900000   // edges + self loops
#define NEG_SLOPE 0.2f
#define GN_EPS    1e-5f

typedef __attribute__((ext_vector_type(2))) float v2f;
typedef __attribute__((ext_vector_type(8))) float v8f;

// ---------- order-preserving float <-> uint for atomic max ----------
__device__ __forceinline__ unsigned fenc(float f) {
    unsigned u = __float_as_uint(f);
    return (u & 0x80000000u) ? ~u : (u | 0x80000000u);
}
__device__ __forceinline__ float fdec(unsigned u) {
    u = (u & 0x80000000u) ? (u ^ 0x80000000u) : ~u;
    return __uint_as_float(u);
}
#define ENC_NEG_INF 0x007FFFFFu   // fenc(-inf)

// ---------- generic fill ----------
__global__ void k_fill_u32(unsigned* __restrict__ p, unsigned v, long long n) {
    long long i = (long long)blockIdx.x * blockDim.x + threadIdx.x;
    long long stride = (long long)gridDim.x * blockDim.x;
    for (; i < n; i += stride) p[i] = v;
}

// ---------- 1) x = X @ W via fp32 WMMA 16x16x4 ----------
// Register-blocked: one wave owns a 32x64 output tile (2 M-tiles x 4 N-tiles,
// 8 v8f accumulators). Per K-step: 2 A + 4 B fragment loads feed 8 WMMAs.
__global__ void k_gemm_wmma(const float* __restrict__ X, const float* __restrict__ W,
                            float* __restrict__ Xp) {
    const int lane = threadIdx.x & 31;
    const int wave = blockIdx.x * (blockDim.x >> 5) + (threadIdx.x >> 5);
    const int tile_n4 = wave & 3;        // 256/64 = 4 column groups
    const int tile_m2 = wave >> 2;       // 100000/32 = 3125 row groups, exact
    if (tile_m2 >= NN / 32) return;

    const int half = lane >> 4;          // which K-half this lane feeds
    const int l16  = lane & 15;
    const int arow0 = tile_m2 * 32 + l16;       // A M-tile 0 (lanes 0..15 = rows)
    const int arow1 = arow0 + 16;               // A M-tile 1
    const int bcol0 = tile_n4 * 64 + l16;       // B/D columns (lanes = cols)

    const float* __restrict__ a0p = X + (long long)arow0 * FIN;
    const float* __restrict__ a1p = X + (long long)arow1 * FIN;

    v8f acc[2][4] = {};
#pragma unroll 2
    for (int k0 = 0; k0 < FIN; k0 += 4) {
        const int ka = k0 + half * 2;
        v2f a0 = *(const v2f*)(a0p + ka);       // A[M][ka], A[M][ka+1]
        v2f a1 = *(const v2f*)(a1p + ka);
        const float* __restrict__ wr0 = W + (long long)ka * HC + bcol0;
        const float* __restrict__ wr1 = wr0 + HC;
        v2f b[4];
#pragma unroll
        for (int j = 0; j < 4; ++j) {           // B[K=ka..ka+1][N = bcol0 + 16j]
            b[j].x = wr0[j * 16];
            b[j].y = wr1[j * 16];
        }
#pragma unroll
        for (int j = 0; j < 4; ++j) {
            acc[0][j] = __builtin_amdgcn_wmma_f32_16x16x4_f32(
                false, a0, false, b[j], (short)0, acc[0][j], false, false);
            acc[1][j] = __builtin_amdgcn_wmma_f32_16x16x4_f32(
                false, a1, false, b[j], (short)0, acc[1][j], false, false);
        }
    }
#pragma unroll
    for (int m = 0; m < 2; ++m) {
#pragma unroll
        for (int j = 0; j < 4; ++j) {
#pragma unroll
            for (int i = 0; i < 8; ++i) {       // D: vgpr i -> M = i (+8 for hi half)
                const int row = tile_m2 * 32 + m * 16 + i + half * 8;
                Xp[(long long)row * HC + bcol0 + j * 16] = acc[m][j][i];
            }
        }
    }
}

// ---------- 2) per-node attention coefficients a_src/a_dst [N,H] ----------
__global__ void k_attn_coef(const float* __restrict__ Xp,
                            const float* __restrict__ att_src,
                            const float* __restrict__ att_dst,
                            float* __restrict__ a_src, float* __restrict__ a_dst) {
    int t = blockIdx.x * blockDim.x + threadIdx.x;     // t = n*H + h
    if (t >= NN * HH) return;
    const int n = t >> 2, h = t & 3;
    const float4* xv = (const float4*)(Xp + (long long)n * HC + h * CC);
    const float4* as = (const float4*)(att_src + h * CC);
    const float4* ad = (const float4*)(att_dst + h * CC);
    float ssum = 0.f, dsum = 0.f;
#pragma unroll
    for (int i = 0; i < CC / 4; ++i) {
        float4 x4 = xv[i], s4 = as[i], d4 = ad[i];
        ssum += x4.x * s4.x + x4.y * s4.y + x4.z * s4.z + x4.w * s4.w;
        dsum += x4.x * d4.x + x4.y * d4.y + x4.z * d4.z + x4.w * d4.w;
    }
    a_src[t] = ssum;
    a_dst[t] = dsum;
}

__device__ __forceinline__ void edge_sd(const long long* __restrict__ ei, int e,
                                        int& s, int& d) {
    if (e < EE) { s = (int)ei[e]; d = (int)ei[EE + e]; }
    else        { s = d = e - EE; }                    // self loops
}

// ---------- 3) segment max of leaky_relu(a_src[s]+a_dst[d]) ----------
__global__ void k_edge_max(const long long* __restrict__ ei,
                           const float* __restrict__ a_src,
                           const float* __restrict__ a_dst,
                           unsigned* __restrict__ amax) {
    int e = blockIdx.x * blockDim.x + threadIdx.x;
    if (e >= EF) return;
    int s, d; edge_sd(ei, e, s, d);
    float4 as = *(const float4*)(a_src + (long long)s * HH);
    float4 ad = *(const float4*)(a_dst + (long long)d * HH);
    float al[4] = {as.x + ad.x, as.y + ad.y, as.z + ad.z, as.w + ad.w};
#pragma unroll
    for (int h = 0; h < HH; ++h) {
        float v = al[h];
        v = v > 0.f ? v : NEG_SLOPE * v;
        atomicMax(&amax[(long long)d * HH + h], fenc(v));
    }
}

// ---------- 4) ex = exp(alpha - amax[dst]); denom[dst] += ex ----------
__global__ void k_edge_exp(const long long* __restrict__ ei,
                           const float* __restrict__ a_src,
                           const float* __restrict__ a_dst,
                           const unsigned* __restrict__ amax,
                           float* __restrict__ ex, float* __restrict__ denom) {
    int e = blockIdx.x * blockDim.x + threadIdx.x;
    if (e >= EF) return;
    int s, d; edge_sd(ei, e, s, d);
    float4 as = *(const float4*)(a_src + (long long)s * HH);
    float4 ad = *(const float4*)(a_dst + (long long)d * HH);
    float al[4] = {as.x + ad.x, as.y + ad.y, as.z + ad.z, as.w + ad.w};
    float4 exo;
    float* exp_o = &exo.x;
#pragma unroll
    for (int h = 0; h < HH; ++h) {
        float v = al[h];
        v = v > 0.f ? v : NEG_SLOPE * v;
        float m = fdec(amax[(long long)d * HH + h]);   // finite: self-loop guarantees
        float xv = __expf(v - m);
        exp_o[h] = xv;
        atomicAdd(&denom[(long long)d * HH + h], xv);
    }
    *(float4*)(ex + (long long)e * HH) = exo;
}

// ---------- 5) scatter: accum[d][c] += sum_h attn_h * x[s][h][c]  (wave/edge) ----------
__global__ void k_edge_scatter(const long long* __restrict__ ei,
                               const float* __restrict__ Xp,
                               const float* __restrict__ ex,
                               const float* __restrict__ denom,
                               float* __restrict__ accum) {
    const int lane = threadIdx.x & 31;
    const int e = blockIdx.x * (blockDim.x >> 5) + (threadIdx.x >> 5);
    if (e >= EF) return;
    int s, d; edge_sd(ei, e, s, d);
    float4 ex4 = *(const float4*)(ex + (long long)e * HH);
    float4 dn4 = *(const float4*)(denom + (long long)d * HH);
    float attn[4] = {ex4.x / (dn4.x + 1e-16f), ex4.y / (dn4.y + 1e-16f),
                     ex4.z / (dn4.z + 1e-16f), ex4.w / (dn4.w + 1e-16f)};
    const float* __restrict__ xs = Xp + (long long)s * HC;
    const int c0 = lane * 2;                           // 2 consecutive channels / lane
    float v0 = 0.f, v1 = 0.f;
#pragma unroll
    for (int h = 0; h < HH; ++h) {
        float2 xv = *(const float2*)(xs + h * CC + c0);
        v0 += attn[h] * xv.x;
        v1 += attn[h] * xv.y;
    }
    float* acc = accum + (long long)d * CC + c0;
    atomicAdd(acc, v0);
    atomicAdd(acc + 1, v1);
}

// ---------- 6) out = accum/H + bias; block-reduced moments ----------
__global__ void k_out_stats(const float* __restrict__ accum,
                            const float* __restrict__ bias,
                            float* __restrict__ outbuf, float* __restrict__ stats) {
    __shared__ float ss[256], sq[256];
    const int c = threadIdx.x & 63, slot = threadIdx.x >> 6;
    const int base = blockIdx.x * 128;
    const float b = bias[c];
    float s1 = 0.f, s2 = 0.f;
    for (int r = slot; r < 128; r += 4) {
        const int n = base + r;
        if (n < NN) {
            float v = accum[(long long)n * CC + c] * (1.0f / HH) + b;
            outbuf[(long long)n * CC + c] = v;
            s1 += v; s2 += v * v;
        }
    }
    ss[threadIdx.x] = s1; sq[threadIdx.x] = s2;
    __syncthreads();
    if (slot == 0) {
#pragma unroll
        for (int j = 1; j < 4; ++j) { s1 += ss[c + 64 * j]; s2 += sq[c + 64 * j]; }
        atomicAdd(&stats[c], s1);
        atomicAdd(&stats[CC + c], s2);
    }
}

// ---------- 7) GraphNorm in place on d_out ----------
__global__ void k_graphnorm(float* __restrict__ outbuf, const float* __restrict__ stats,
                            const float* __restrict__ gw, const float* __restrict__ gb,
                            const float* __restrict__ gms) {
    long long i = (long long)blockIdx.x * blockDim.x + threadIdx.x;
    if (i >= (long long)NN * CC) return;
    const int c = (int)(i & 63);
    const float mu  = stats[c] * (1.0f / NN);
    const float eo2 = stats[CC + c] * (1.0f / NN);
    const float s   = gms[c];
    const float var = eo2 - 2.f * s * mu * mu + s * s * mu * mu;  // E[(o - s*mu)^2]
    const float cen = outbuf[i] - s * mu;
    outbuf[i] = gw[c] * cen * __frsqrt_rn(var + GN_EPS) + gb[c];
}

extern "C" void kernel_launch(void* const* d_in, const int* in_sizes, int n_in,
                              void* d_out, int out_size, void* d_ws, size_t ws_size,
                              hipStream_t stream) {
    const float*     X        = (const float*)d_in[0];
    const long long* ei       = (const long long*)d_in[1];   // int64 [2,E]
    // d_in[2] = attr (pass-through in reference, unused)
    const float*     W        = (const float*)d_in[3];
    const float*     att_src  = (const float*)d_in[4];
    const float*     att_dst  = (const float*)d_in[5];
    const float*     bias     = (const float*)d_in[6];
    const float*     gw       = (const float*)d_in[7];
    const float*     gb       = (const float*)d_in[8];
    const float*     gms      = (const float*)d_in[9];
    float* out = (float*)d_out;

    // workspace layout (floats)
    float* wsf = (float*)d_ws;
    float*    Xp    = wsf;                                 // N*256
    float*    a_src = Xp    + (long long)NN * HC;          // N*4
    float*    a_dst = a_src + (long long)NN * HH;          // N*4
    float*    ex    = a_dst + (long long)NN * HH;          // EF*4
    unsigned* amax  = (unsigned*)(ex + (long long)EF * HH);// N*4
    float*    denom = (float*)(amax + (long long)NN * HH); // N*4
    float*    accum = denom + (long long)NN * HH;          // N*64
    float*    stats = accum + (long long)NN * CC;          // 128

    // init (every call: deterministic, graph-capture safe)
    {
        long long n_amax = (long long)NN * HH;
        long long n_zero = (long long)NN * HH + (long long)NN * CC + 2 * CC; // denom|accum|stats
        k_fill_u32<<<1024, 256, 0, stream>>>(amax, ENC_NEG_INF, n_amax);
        k_fill_u32<<<4096, 256, 0, stream>>>((unsigned*)denom, 0u, n_zero);
    }

    // 1) projection GEMM (WMMA f32): 3125*4 waves, each owns a 32x64 tile
    {
        int waves = (NN / 32) * (HC / 64);
        k_gemm_wmma<<<(waves + 7) / 8, 256, 0, stream>>>(X, W, Xp);
    }
    // 2) attention coefficients
    k_attn_coef<<<(NN * HH + 255) / 256, 256, 0, stream>>>(Xp, att_src, att_dst, a_src, a_dst);
    // 3) segment max
    k_edge_max<<<(EF + 255) / 256, 256, 0, stream>>>(ei, a_src, a_dst, amax);
    // 4) exp + denom
    k_edge_exp<<<(EF + 255) / 256, 256, 0, stream>>>(ei, a_src, a_dst, amax, ex, denom);
    // 5) weighted scatter (wave per edge, 8 waves/block)
    k_edge_scatter<<<(EF + 7) / 8, 256, 0, stream>>>(ei, Xp, ex, denom, accum);
    // 6) head-mean + bias + moments
    k_out_stats<<<(NN + 127) / 128, 256, 0, stream>>>(accum, bias, out, stats);
    // 7) normalize
    k_graphnorm<<<(int)(((long long)NN * CC + 255) / 256), 256, 0, stream>>>(out, stats, gw, gb, gms);

    (void)in_sizes; (void)n_in; (void)out_size; (void)ws_size;
}